// LocalAggregator_62921270886588
// MI455X (gfx1250) — compile-verified
//
#include <hip/hip_runtime.h>

// ---------------- constants ----------------
#define Bc    64
#define NMAXc 256
#define Tc    16
#define Hc    32
#define Nn_   16384      // B*NMAX
#define Ec    131072     // edges
#define EPSc  1e-15f

typedef float v2f __attribute__((ext_vector_type(2)));
typedef float v8f __attribute__((ext_vector_type(8)));

// ---------------- WMMA f32 16x16x4 helpers ----------------
__device__ __forceinline__ v8f wmma4(v2f a, v2f b, v8f c) {
  return __builtin_amdgcn_wmma_f32_16x16x4_f32(false, a, false, b, (short)0, c,
                                               false, false);
}

// C(16x16) += A(16xK, row-major, lda) * B(Kx16, row-major, ldb)
__device__ __forceinline__ v8f wmma_tile(const float* __restrict__ A, int lda,
                                         const float* __restrict__ Bm, int ldb,
                                         int K, v8f acc) {
  const int lane = threadIdx.x & 31;
  const int l15  = lane & 15;
  const int kb   = (lane >> 4) << 1;   // 0 or 2
  for (int k0 = 0; k0 < K; k0 += 4) {
    int ka = k0 + kb;
    v2f a, b;
    a.x = A[l15 * lda + ka];
    a.y = A[l15 * lda + ka + 1];
    b.x = Bm[ka * ldb + l15];
    b.y = Bm[(ka + 1) * ldb + l15];
    acc = wmma4(a, b, acc);
  }
  return acc;
}

// C(16x16) += (At^T)(16xK) * B(Kx16);  At stored [K x 16] row-major (ldat)
__device__ __forceinline__ v8f wmma_tile_At(const float* __restrict__ At, int ldat,
                                            const float* __restrict__ Bm, int ldb,
                                            int K, v8f acc) {
  const int lane = threadIdx.x & 31;
  const int l15  = lane & 15;
  const int kb   = (lane >> 4) << 1;
  for (int k0 = 0; k0 < K; k0 += 4) {
    int ka = k0 + kb;
    v2f a, b;
    a.x = At[ka * ldat + l15];
    a.y = At[(ka + 1) * ldat + l15];
    b.x = Bm[ka * ldb + l15];
    b.y = Bm[(ka + 1) * ldb + l15];
    acc = wmma4(a, b, acc);
  }
  return acc;
}

__device__ __forceinline__ void store_tile(float* __restrict__ C, int ldc, v8f acc,
                                           const float* __restrict__ bias, int accum) {
  const int lane  = threadIdx.x & 31;
  const int l15   = lane & 15;
  const int rbase = (lane >> 4) << 3;  // 0 or 8
  float bv = bias ? bias[l15] : 0.f;
#pragma unroll
  for (int r = 0; r < 8; ++r) {
    float v  = acc[r] + bv;
    float* p = C + (long)(r + rbase) * ldc + l15;
    if (accum) *p += v; else *p = v;
  }
}

// ---------------- generic batched WMMA GEMM ----------------
// C_b[M x Nc] = A_b[M x K] @ B_b[K x Nc] (+bias[Nc]) (+=)
__global__ void __launch_bounds__(32)
k_bgemm(const float* __restrict__ A, const float* __restrict__ Bm,
        float* __restrict__ C, const float* __restrict__ bias,
        int K, int ldb, int ldc,
        long aBS, long bBS, long cBS, int aDiv, int accum) {
  int batch = blockIdx.x;
  const float* Ab = A + (long)(batch / aDiv) * aBS;
  const float* Bb = Bm + (long)batch * bBS;
  float* Cb = C + (long)batch * cBS;
  int tm = blockIdx.y << 4, tn = blockIdx.z << 4;
  v8f acc = {};
  acc = wmma_tile(Ab + (long)tm * K, K, Bb + tn, ldb, K, acc);
  store_tile(Cb + (long)tm * ldc + tn, ldc, acc, bias ? bias + tn : nullptr, accum);
}

// C_b[M x Nc] = (At_b)^T @ B_b;  At stored [K x M] (ldat)
__global__ void __launch_bounds__(32)
k_bgemm_At(const float* __restrict__ At, const float* __restrict__ Bm,
           float* __restrict__ C,
           int K, int ldat, int ldb, int ldc,
           long aBS, long bBSb, long bBSt, int tDiv, long cBS, int aDiv) {
  int batch = blockIdx.x;
  const float* Ab = At + (long)(batch / aDiv) * aBS;
  const float* Bb = Bm + (long)(batch / tDiv) * bBSb + (long)(batch % tDiv) * bBSt;
  float* Cb = C + (long)batch * cBS;
  int tm = blockIdx.y << 4, tn = blockIdx.z << 4;
  v8f acc = {};
  acc = wmma_tile_At(Ab + tm, ldat, Bb + tn, ldb, K, acc);
  store_tile(Cb + (long)tm * ldc + tn, ldc, acc, nullptr, 0);
}

// ---------------- small helper kernels ----------------
__global__ void k_fill(float* p, float v, long n) {
  long i = (long)blockIdx.x * blockDim.x + threadIdx.x;
  if (i < n) p[i] = v;
}

__global__ void k_deg_adj(const int* __restrict__ esrc, const int* __restrict__ edst,
                          const float* __restrict__ ew, float* __restrict__ deg,
                          float* __restrict__ adj) {
  int i = blockIdx.x * 256 + threadIdx.x;
  if (i >= Ec) return;
  int s = esrc[i], d = edst[i];
  atomicAdd(&deg[d], ew[i]);
  int b = s >> 8;
  atomicAdd(&adj[(long)b * 65536 + (long)(s & 255) * 256 + (d & 255)], 1.0f);
}

__global__ void k_dinv(float* deg, int n) {
  int i = blockIdx.x * 256 + threadIdx.x;
  if (i < n) { float d = deg[i]; deg[i] = d > 0.f ? rsqrtf(d) : 0.f; }
}

// one block per edge: scatter xw[src]*norm into X[dst]
__global__ void __launch_bounds__(512)
k_edge_scatter(const float* __restrict__ pos, const float* __restrict__ W1,
               const float* __restrict__ ew, const int* __restrict__ esrc,
               const int* __restrict__ edst, const float* __restrict__ dinv,
               float* __restrict__ X) {
  int e = blockIdx.x, tid = threadIdx.x;
  __shared__ float sp[48];
  __shared__ float snrm;
  int s = esrc[e], dn = edst[e];
  if (tid < 48) sp[tid] = pos[(long)s * 48 + tid];
  if (tid == 0) snrm = dinv[s] * ew[e] * dinv[dn];
  __syncthreads();
  int t = tid >> 5, h = tid & 31;
  float v = (sp[t*3+0]*W1[h] + sp[t*3+1]*W1[Hc+h] + sp[t*3+2]*W1[2*Hc+h]) * snrm;
  atomicAdd(&X[((long)dn * Tc + t) * Hc + h], v);
}

__global__ void k_x_final(const float* __restrict__ pos, const float* __restrict__ W1,
                          const float* __restrict__ b1, const float* __restrict__ dinv,
                          float* __restrict__ X) {
  long i = (long)blockIdx.x * 256 + threadIdx.x;   // < N*T*H
  int h = i & 31; int t = (i >> 5) & 15; long n = i >> 9;
  float dv = dinv[n];
  const float* p = pos + n * 48 + t * 3;
  float self = (p[0]*W1[h] + p[1]*W1[Hc+h] + p[2]*W1[2*Hc+h]) * dv * dv;
  float v = X[i] + self + b1[h];
  X[i] = v > 0.f ? v : 0.f;
}

// out[r,h] = mean_t in[(r/div)*oStr + (r%div)*iStr + t*tStr + h]
__global__ void k_meanT(const float* __restrict__ in, float* __restrict__ out,
                        long rows, long div, long oStr, long iStr, long tStr) {
  long idx = (long)blockIdx.x * 256 + threadIdx.x;
  if (idx >= rows * 32) return;
  long r = idx >> 5; int h = idx & 31;
  const float* p = in + (r / div) * oStr + (r % div) * iStr + h;
  float s = 0.f;
  for (int t = 0; t < Tc; ++t) s += p[(long)t * tStr];
  out[idx] = s * (1.0f / Tc);
}

// one wave per row: tanh then softmax over first V cols, zero cols [V,W)
__global__ void __launch_bounds__(32)
k_tanh_softmax(float* buf, int W, int V) {
  long row = blockIdx.x;
  float* p = buf + row * (long)W;
  int lane = threadIdx.x;
  int i0 = lane, i1 = lane + 32;
  float t0 = 0.f, t1 = 0.f, v0 = -1e30f, v1 = -1e30f;
  if (i0 < V) { t0 = tanhf(p[i0]); v0 = t0; }
  if (i1 < V) { t1 = tanhf(p[i1]); v1 = t1; }
  float m = fmaxf(v0, v1);
  for (int o = 16; o; o >>= 1) m = fmaxf(m, __shfl_xor(m, o, 32));
  float e0 = (i0 < V) ? __expf(t0 - m) : 0.f;
  float e1 = (i1 < V) ? __expf(t1 - m) : 0.f;
  float s = e0 + e1;
  for (int o = 16; o; o >>= 1) s += __shfl_xor(s, o, 32);
  float inv = 1.f / s;
  if (i0 < W) p[i0] = (i0 < V) ? e0 * inv : 0.f;
  if (i1 < W) p[i1] = (i1 < V) ? e1 * inv : 0.f;
}

// per-graph mincut ratio: trace(oadj) / sum_n d[n]*sum_k s[n,k]^2
__global__ void __launch_bounds__(256)
k_mincut(const float* __restrict__ adjn, long adjBS, int Nn,
         const float* __restrict__ oadj, long oadjBS, int ldo,
         const float* __restrict__ P, long pBS, int ldp, int Kc,
         float* __restrict__ out) {
  int b = blockIdx.x, tid = threadIdx.x;
  const float* A = adjn + (long)b * adjBS;
  const float* O = oadj + (long)b * oadjBS;
  const float* S = P + (long)b * pBS;
  __shared__ float red[256];
  float num = 0.f;
  for (int k = tid; k < Kc; k += 256) num += O[(long)k * ldo + k];
  float den = 0.f;
  for (int n = tid; n < Nn; n += 256) {
    float d = 0.f;
    for (int m = 0; m < Nn; ++m) d += A[(long)n * Nn + m];
    float ss = 0.f;
    for (int k = 0; k < Kc; ++k) { float v = S[(long)n * ldp + k]; ss += v * v; }
    den += d * ss;
  }
  red[tid] = num; __syncthreads();
  for (int s = 128; s; s >>= 1) { if (tid < s) red[tid] += red[tid + s]; __syncthreads(); }
  float numT = red[0]; __syncthreads();
  red[tid] = den; __syncthreads();
  for (int s = 128; s; s >>= 1) { if (tid < s) red[tid] += red[tid + s]; __syncthreads(); }
  if (tid == 0) out[b] = numT / red[0];
}

// per-graph ortho loss: || SS/||SS||_F - I/sqrt(Kc) ||_F,  SS = S^T S
__global__ void __launch_bounds__(256)
k_ortho(const float* __restrict__ P, long pBS, int ldp, int Nn, int Kc,
        float* __restrict__ out) {
  int b = blockIdx.x, tid = threadIdx.x;
  const float* S = P + (long)b * pBS;
  __shared__ float ss[64 * 64];
  __shared__ float red[256];
  int KK = Kc * Kc;
  for (int idx = tid; idx < KK; idx += 256) {
    int k = idx / Kc, l = idx % Kc;
    float a = 0.f;
    for (int n = 0; n < Nn; ++n) a += S[(long)n * ldp + k] * S[(long)n * ldp + l];
    ss[idx] = a;
  }
  __syncthreads();
  float sq = 0.f;
  for (int idx = tid; idx < KK; idx += 256) sq += ss[idx] * ss[idx];
  red[tid] = sq; __syncthreads();
  for (int s = 128; s; s >>= 1) { if (tid < s) red[tid] += red[tid + s]; __syncthreads(); }
  float nrm = sqrtf(red[0]); __syncthreads();
  float isq = rsqrtf((float)Kc);
  float acc = 0.f;
  for (int idx = tid; idx < KK; idx += 256) {
    int k = idx / Kc, l = idx % Kc;
    float v = ss[idx] / nrm - ((k == l) ? isq : 0.f);
    acc += v * v;
  }
  red[tid] = acc; __syncthreads();
  for (int s = 128; s; s >>= 1) { if (tid < s) red[tid] += red[tid + s]; __syncthreads(); }
  if (tid == 0) out[b] = sqrtf(red[0]);
}

// zero diag, d = sqrt(rowsum)+EPS, oadj[k,l] /= d[k]*d[l]   (in place)
__global__ void __launch_bounds__(256)
k_adj_norm(float* __restrict__ oadj, long BS, int Kc) {
  int b = blockIdx.x, tid = threadIdx.x;
  float* O = oadj + (long)b * BS;
  __shared__ float m[64 * 64];
  __shared__ float dr[64];
  int KK = Kc * Kc;
  for (int idx = tid; idx < KK; idx += 256) {
    int k = idx / Kc, l = idx % Kc;
    m[idx] = (k == l) ? 0.f : O[idx];
  }
  __syncthreads();
  if (tid < Kc) {
    float s = 0.f;
    for (int l = 0; l < Kc; ++l) s += m[tid * Kc + l];
    dr[tid] = sqrtf(s) + EPSc;
  }
  __syncthreads();
  for (int idx = tid; idx < KK; idx += 256) {
    int k = idx / Kc, l = idx % Kc;
    O[idx] = m[idx] / (dr[k] * dr[l]);
  }
}

__global__ void k_pad3(const float* __restrict__ Wp3, const float* __restrict__ bp3,
                       float* __restrict__ wout, float* __restrict__ bout) {
  int i = blockIdx.x * 256 + threadIdx.x;
  if (i < 32 * 16) {
    int r = i >> 4, c = i & 15;
    wout[i] = (c < 8) ? Wp3[r * 8 + c] : 0.f;
  } else if (i < 32 * 16 + 16) {
    int c = i - 512;
    bout[c] = (c < 8) ? bp3[c] : 0.f;
  }
}

__global__ void k_pack_x3(const float* __restrict__ xin, float* __restrict__ out) {
  long i = (long)blockIdx.x * 256 + threadIdx.x;  // B*T*8*64
  if (i >= (long)Bc * Tc * 8 * 64) return;
  int o = i & 63; int k = (i >> 6) & 7; long bt = i >> 9;
  out[i] = xin[(bt * 16 + k) * 64 + o];
}

__global__ void k_pack_agg(const float* __restrict__ ain, float* __restrict__ out) {
  long i = (long)blockIdx.x * 256 + threadIdx.x;  // B*256*8
  if (i >= (long)Bc * 256 * 8) return;
  int k = i & 7; long n = i >> 3;
  out[i] = ain[n * 16 + k];
}

__global__ void __launch_bounds__(64)
k_loss_final(const float* __restrict__ mcp, const float* __restrict__ orp,
             float* __restrict__ out_mc, float* __restrict__ out_or) {
  __shared__ float r1[64], r2[64];
  int tid = threadIdx.x;
  float a = 0.f, c = 0.f;
  for (int i = tid; i < 3 * Bc; i += 64) { a += mcp[i]; c += orp[i]; }
  r1[tid] = a; r2[tid] = c; __syncthreads();
  for (int s = 32; s; s >>= 1) {
    if (tid < s) { r1[tid] += r1[tid + s]; r2[tid] += r2[tid + s]; }
    __syncthreads();
  }
  if (tid == 0) { *out_mc = -r1[0] / (float)Bc; *out_or = r2[0] / (float)Bc; }
}

// ---------------- host launch ----------------
extern "C" void kernel_launch(void* const* d_in, const int* in_sizes, int n_in,
                              void* d_out, int out_size, void* d_ws, size_t ws_size,
                              hipStream_t stream) {
  (void)in_sizes; (void)n_in; (void)out_size; (void)ws_size;
  const float* pos   = (const float*)d_in[0];
  const float* ew    = (const float*)d_in[1];
  const int*   esrc  = (const int*)d_in[2];
  const int*   edst  = (const int*)d_in[3];
  const float* W1    = (const float*)d_in[4];
  const float* b1    = (const float*)d_in[5];
  const float* Wp1   = (const float*)d_in[6];
  const float* bp1   = (const float*)d_in[7];
  const float* Wrel2 = (const float*)d_in[8];
  const float* brel2 = (const float*)d_in[9];
  const float* Wroot2= (const float*)d_in[10];
  const float* Wp2   = (const float*)d_in[11];
  const float* bp2   = (const float*)d_in[12];
  const float* Wrel3 = (const float*)d_in[13];
  const float* brel3 = (const float*)d_in[14];
  const float* Wroot3= (const float*)d_in[15];
  const float* Wp3   = (const float*)d_in[16];
  const float* bp3   = (const float*)d_in[17];
  const float* Wrel4 = (const float*)d_in[18];
  const float* brel4 = (const float*)d_in[19];
  const float* Wroot4= (const float*)d_in[20];
  float* out = (float*)d_out;

  float* ws = (float*)d_ws;
  float* X     = ws;                   // [N,T,H] 8388608
  float* DEG   = X     + 8388608;      // 16384
  float* ADJ   = DEG   + 16384;        // [B,256,256] 4194304
  float* XM    = ADJ   + 4194304;      // mean buffer 524288
  float* P1    = XM    + 524288;       // [B,256,64] 1048576
  float* X1    = P1    + 1048576;      // [B,T,64,32] 2097152
  float* OADJ1 = X1    + 2097152;      // [B,64,64] 262144
  float* X1C   = OADJ1 + 262144;       // 2097152
  float* P2    = X1C   + 2097152;      // [B,64,32] 131072
  float* X2    = P2    + 131072;       // [B,T,32,32] 1048576
  float* OADJ2 = X2    + 1048576;      // [B,32,32] 65536
  float* X2C   = OADJ2 + 65536;        // 1048576
  float* P3    = X2C   + 1048576;      // [B,32,16] 32768 (padded K=8->16)
  float* X3    = P3    + 32768;        // [B,T,16,32] 524288
  float* OADJ3 = X3    + 524288;       // [B,16,16] 16384
  float* X3O   = OADJ3 + 16384;        // [B,T,16,64] 1048576
  float* TMP   = X3O   + 1048576;      // scratch 2097152
  float* AGGP  = TMP   + 2097152;      // [B,256,16] 262144
  float* W3P   = AGGP  + 262144;       // 512
  float* B3P   = W3P   + 512;          // 16
  float* MCP   = B3P   + 16;           // 192
  float* ORP   = MCP   + 192;          // 192

  // ---- GCN layer (sparse) ----
  k_fill<<<32768, 256, 0, stream>>>(X, 0.f, 8388608L);
  k_fill<<<64, 256, 0, stream>>>(DEG, 1.0f, 16384L);
  k_fill<<<16384, 256, 0, stream>>>(ADJ, 0.f, 4194304L);
  k_deg_adj<<<512, 256, 0, stream>>>(esrc, edst, ew, DEG, ADJ);
  k_dinv<<<64, 256, 0, stream>>>(DEG, Nn_);
  k_edge_scatter<<<Ec, 512, 0, stream>>>(pos, W1, ew, esrc, edst, DEG, X);
  k_x_final<<<32768, 256, 0, stream>>>(pos, W1, b1, DEG, X);

  // ---- level 1: assignments ----
  k_meanT<<<2048, 256, 0, stream>>>(X, XM, 16384L, 16384L, 0L, 512L, 32L);
  k_bgemm<<<dim3(1, 1024, 4), 32, 0, stream>>>(XM, Wp1, P1, bp1,
      32, 64, 64, 0L, 0L, 0L, 1, 0);
  k_tanh_softmax<<<16384, 32, 0, stream>>>(P1, 64, 64);
  // x1[b,t,k,h] = P1^T * X   (X layout [b,n,t,h])
  k_bgemm_At<<<dim3(1024, 4, 2), 32, 0, stream>>>(P1, X, X1,
      256, 64, 512, 32, 16384L, 131072L, 32L, 16, 2048L, 16);
  // oadj1 = P1^T (adj P1)
  k_bgemm<<<dim3(64, 16, 4), 32, 0, stream>>>(ADJ, P1, TMP, nullptr,
      256, 64, 64, 65536L, 16384L, 16384L, 1, 0);
  k_bgemm_At<<<dim3(64, 4, 4), 32, 0, stream>>>(P1, TMP, OADJ1,
      256, 64, 64, 64, 16384L, 16384L, 0L, 1, 4096L, 1);
  k_mincut<<<64, 256, 0, stream>>>(ADJ, 65536L, 256, OADJ1, 4096L, 64,
      P1, 16384L, 64, 64, MCP + 0);
  k_ortho<<<64, 256, 0, stream>>>(P1, 16384L, 64, 256, 64, ORP + 0);
  k_adj_norm<<<64, 256, 0, stream>>>(OADJ1, 4096L, 64);

  // ---- conv 2: x1c = (adj1 x1) Wrel2 + brel2 + x1 Wroot2 ----
  k_bgemm<<<dim3(1024, 4, 2), 32, 0, stream>>>(OADJ1, X1, TMP, nullptr,
      64, 32, 32, 4096L, 2048L, 2048L, 16, 0);
  k_bgemm<<<dim3(1024, 4, 2), 32, 0, stream>>>(TMP, Wrel2, X1C, brel2,
      32, 32, 32, 2048L, 0L, 2048L, 1, 0);
  k_bgemm<<<dim3(1024, 4, 2), 32, 0, stream>>>(X1, Wroot2, X1C, nullptr,
      32, 32, 32, 2048L, 0L, 2048L, 1, 1);

  // ---- level 2 ----
  k_meanT<<<512, 256, 0, stream>>>(X1C, XM, 4096L, 64L, 32768L, 32L, 2048L);
  k_bgemm<<<dim3(1, 256, 2), 32, 0, stream>>>(XM, Wp2, P2, bp2,
      32, 32, 32, 0L, 0L, 0L, 1, 0);
  k_tanh_softmax<<<4096, 32, 0, stream>>>(P2, 32, 32);
  k_bgemm_At<<<dim3(1024, 2, 2), 32, 0, stream>>>(P2, X1C, X2,
      64, 32, 32, 32, 2048L, 2048L, 0L, 1, 1024L, 16);
  k_bgemm<<<dim3(64, 4, 2), 32, 0, stream>>>(OADJ1, P2, TMP, nullptr,
      64, 32, 32, 4096L, 2048L, 2048L, 1, 0);
  k_bgemm_At<<<dim3(64, 2, 2), 32, 0, stream>>>(P2, TMP, OADJ2,
      64, 32, 32, 32, 2048L, 2048L, 0L, 1, 1024L, 1);
  k_mincut<<<64, 256, 0, stream>>>(OADJ1, 4096L, 64, OADJ2, 1024L, 32,
      P2, 2048L, 32, 32, MCP + 64);
  k_ortho<<<64, 256, 0, stream>>>(P2, 2048L, 32, 64, 32, ORP + 64);
  k_adj_norm<<<64, 256, 0, stream>>>(OADJ2, 1024L, 32);

  // ---- conv 3 ----
  k_bgemm<<<dim3(1024, 2, 2), 32, 0, stream>>>(OADJ2, X2, TMP, nullptr,
      32, 32, 32, 1024L, 1024L, 1024L, 16, 0);
  k_bgemm<<<dim3(1024, 2, 2), 32, 0, stream>>>(TMP, Wrel3, X2C, brel3,
      32, 32, 32, 1024L, 0L, 1024L, 1, 0);
  k_bgemm<<<dim3(1024, 2, 2), 32, 0, stream>>>(X2, Wroot3, X2C, nullptr,
      32, 32, 32, 1024L, 0L, 1024L, 1, 1);

  // ---- level 3 (K=8 padded to 16) ----
  k_pad3<<<3, 256, 0, stream>>>(Wp3, bp3, W3P, B3P);
  k_meanT<<<256, 256, 0, stream>>>(X2C, XM, 2048L, 32L, 16384L, 32L, 1024L);
  k_bgemm<<<dim3(1, 128, 1), 32, 0, stream>>>(XM, W3P, P3, B3P,
      32, 16, 16, 0L, 0L, 0L, 1, 0);
  k_tanh_softmax<<<2048, 32, 0, stream>>>(P3, 16, 8);
  k_bgemm_At<<<dim3(1024, 1, 2), 32, 0, stream>>>(P3, X2C, X3,
      32, 16, 32, 32, 512L, 1024L, 0L, 1, 512L, 16);
  k_bgemm<<<dim3(64, 2, 1), 32, 0, stream>>>(OADJ2, P3, TMP, nullptr,
      32, 16, 16, 1024L, 512L, 512L, 1, 0);
  k_bgemm_At<<<dim3(64, 1, 1), 32, 0, stream>>>(P3, TMP, OADJ3,
      32, 16, 16, 16, 512L, 512L, 0L, 1, 256L, 1);
  k_mincut<<<64, 256, 0, stream>>>(OADJ2, 1024L, 32, OADJ3, 256L, 16,
      P3, 512L, 16, 8, MCP + 128);
  k_ortho<<<64, 256, 0, stream>>>(P3, 512L, 16, 32, 8, ORP + 128);
  k_adj_norm<<<64, 256, 0, stream>>>(OADJ3, 256L, 16);

  // ---- conv 4 (final) ----
  k_bgemm<<<dim3(1024, 1, 2), 32, 0, stream>>>(OADJ3, X3, TMP, nullptr,
      16, 32, 32, 256L, 512L, 512L, 16, 0);
  k_bgemm<<<dim3(1024, 1, 4), 32, 0, stream>>>(TMP, Wrel4, X3O, brel4,
      32, 64, 64, 512L, 0L, 1024L, 1, 0);
  k_bgemm<<<dim3(1024, 1, 4), 32, 0, stream>>>(X3, Wroot4, X3O, nullptr,
      32, 64, 64, 512L, 0L, 1024L, 1, 1);
  k_pack_x3<<<2048, 256, 0, stream>>>(X3O, out);

  // ---- agg = p1 @ p2 @ p3 ----
  k_bgemm<<<dim3(64, 16, 2), 32, 0, stream>>>(P1, P2, TMP, nullptr,
      64, 32, 32, 16384L, 2048L, 8192L, 1, 0);
  k_bgemm<<<dim3(64, 16, 1), 32, 0, stream>>>(TMP, P3, AGGP, nullptr,
      32, 16, 16, 8192L, 512L, 4096L, 1, 0);
  k_pack_agg<<<512, 256, 0, stream>>>(AGGP, out + 524290);

  // ---- losses ----
  k_loss_final<<<1, 64, 0, stream>>>(MCP, ORP, out + 524288, out + 524289);
}